// TotalLoss_80066780332724
// MI455X (gfx1250) — compile-verified
//
#include <hip/hip_runtime.h>
#include <hip/hip_bf16.h>
#include <math.h>

typedef __attribute__((ext_vector_type(2))) float v2f;
typedef __attribute__((ext_vector_type(8))) float v8f;

#define GAMMA 0.5f
#define ALPHA 0.5f
#define BETA  1.0f
#define THETA 1.0f
#define BB 4
#define NN 16384
#define MM 2048

// ds_swizzle group-of-32 encoding: offset[15]=0, and_mask=0x1f, xor_mask<<10
#define SWZ_XOR(x) (((x) << 10) | 0x1f)

template <int OFFS>
__device__ __forceinline__ float swz_min(float v) {
    int o = __builtin_amdgcn_ds_swizzle(__float_as_int(v), OFFS);
    return fminf(v, __int_as_float(o));
}
template <int OFFS>
__device__ __forceinline__ float swz_add(float v) {
    int o = __builtin_amdgcn_ds_swizzle(__float_as_int(v), OFFS);
    return v + __int_as_float(o);
}

// Workspace accumulator layout (floats):
// [0] sum d_smp_org   [1..4] per-batch max d_smp_org (uint bits)
// [5] sum d_org_smp   [6] sum trans*w  [7] sum rot*w  [8] sum bce
__global__ void init_acc_kernel(float* acc) {
    int i = threadIdx.x;
    if (i < 16) acc[i] = 0.0f;
}

// One wave per 16-row tile of p1; stream all 16-col tiles of p2,
// 4x unrolled: four V_WMMA_F32_16X16X4_F32 per backedge (K=4, z-padded).
__global__ __launch_bounds__(256) void chamfer_min_kernel(
    const float* __restrict__ p1, const float* __restrict__ p2,
    int n1, int n2, int totalTiles,
    float* __restrict__ sumAcc, unsigned* __restrict__ maxAcc) {
    const int lane = threadIdx.x & 31;
    const int wave = (blockIdx.x * blockDim.x + threadIdx.x) >> 5;
    if (wave >= totalTiles) return;
    const int tilesPerBatch = n1 >> 4;
    const int b = wave / tilesPerBatch;
    const int rowBase = (wave - b * tilesPerBatch) << 4;

    // A operand: lanes 0-15 -> (x,y) of row M=rowBase+lane (K=0,1)
    //            lanes 16-31 -> (z,0) of row M=rowBase+lane-16 (K=2,3)
    v2f a;
    {
        int m = rowBase + (lane & 15);
        const float* p = p1 + ((size_t)b * n1 + m) * 3;
        float px = p[0], py = p[1], pz = p[2];
        if (lane < 16) { a.x = px;  a.y = py;   }
        else           { a.x = pz;  a.y = 0.0f; }
    }
    // |p1|^2 for the 8 C rows this lane owns (C: M = r + 8*(lane>=16))
    float p1sq[8];
    {
        int mBase = rowBase + ((lane >> 4) << 3);
        #pragma unroll
        for (int r = 0; r < 8; ++r) {
            const float* p = p1 + ((size_t)b * n1 + mBase + r) * 3;
            float px = p[0], py = p[1], pz = p[2];
            p1sq[r] = px * px + py * py + pz * pz;
        }
    }

    float minD[8];
    #pragma unroll
    for (int r = 0; r < 8; ++r) minD[r] = 3.0e38f;

    const int colTiles = n2 >> 4;          // 1024 or 128, both divisible by 4
    const float* p2base = p2 + (size_t)b * n2 * 3;

    for (int ct = 0; ct < colTiles; ct += 4) {
        // branchless prefetch two unroll-groups ahead (clamped, speculative)
        int pf = ct + 8;
        pf = (pf < colTiles) ? pf : 0;
        __builtin_prefetch(p2base + ((size_t)(pf << 4) + (lane & 15)) * 3, 0, 1);

        #pragma unroll
        for (int u = 0; u < 4; ++u) {
            int nIdx = ((ct + u) << 4) + (lane & 15);
            const float* q = p2base + (size_t)nIdx * 3;
            float qx = q[0], qy = q[1], qz = q[2];
            float qsq = qx * qx + qy * qy + qz * qz;
            v2f bv;
            if (lane < 16) { bv.x = qx; bv.y = qy;   }
            else           { bv.x = qz; bv.y = 0.0f; }

            v8f c = {};
            // 16x16 tile of p1.p2 dot products in one instruction
            c = __builtin_amdgcn_wmma_f32_16x16x4_f32(
                    /*neg_a=*/false, a, /*neg_b=*/false, bv,
                    /*c_mod=*/(short)0, c, /*reuse_a=*/false, /*reuse_b=*/false);

            #pragma unroll
            for (int r = 0; r < 8; ++r)
                minD[r] = fminf(minD[r], p1sq[r] + qsq - 2.0f * c[r]);
        }
    }

    // Min across the 16 lanes of each half (reduces over the N dimension);
    // xor masks < 16 never cross the 16-lane halves.
    #pragma unroll
    for (int r = 0; r < 8; ++r) {
        minD[r] = swz_min<SWZ_XOR(1)>(minD[r]);
        minD[r] = swz_min<SWZ_XOR(2)>(minD[r]);
        minD[r] = swz_min<SWZ_XOR(4)>(minD[r]);
        minD[r] = swz_min<SWZ_XOR(8)>(minD[r]);
    }
    if ((lane & 15) == 0) {
        float s = 0.0f, mx = 0.0f; // distances >= 0, so 0-init max is safe
        #pragma unroll
        for (int r = 0; r < 8; ++r) { s += minD[r]; mx = fmaxf(mx, minD[r]); }
        atomicAdd(sumAcc, s);
        if (maxAcc) atomicMax(&maxAcc[b], __float_as_uint(mx));
    }
}

__global__ __launch_bounds__(256) void grasp_cls_kernel(
    const float* __restrict__ grasp_pred, const float* __restrict__ grasp_gt,
    const float* __restrict__ class_pred, const float* __restrict__ class_gt,
    float* __restrict__ acc) {
    int i = blockIdx.x * blockDim.x + threadIdx.x;
    float transv = 0.0f, rotv = 0.0f, bcev = 0.0f;
    if (i < BB * MM) {
        const float* gp = grasp_pred + (size_t)i * 7;
        float cx = gp[0], cy = gp[1], cz = gp[2];
        float qw = gp[3], qx = gp[4], qy = gp[5], qz = gp[6];
        float qn = rsqrtf(qw * qw + qx * qx + qy * qy + qz * qz + 1e-12f);
        qw *= qn; qx *= qn; qy *= qn; qz *= qn;

        const float* g = grasp_gt + (size_t)i * 16;
        float R00 = g[0], R01 = g[1], R02 = g[2],  tx = g[3];
        float R10 = g[4], R11 = g[5], R12 = g[6],  ty = g[7];
        float R20 = g[8], R21 = g[9], R22 = g[10], tz = g[11];
        float tr = R00 + R11 + R22;
        float w  = 0.5f * sqrtf(fmaxf(1.0f + tr, 1e-12f));
        float inv4w = 1.0f / (4.0f * w);
        float gx = (R21 - R12) * inv4w;
        float gy = (R02 - R20) * inv4w;
        float gz = (R10 - R01) * inv4w;

        float cw = class_gt[i];
        float dx = cx - tx + 1e-6f, dy = cy - ty + 1e-6f, dz = cz - tz + 1e-6f;
        transv = sqrtf(dx * dx + dy * dy + dz * dz) * cw;

        float dot = qw * w + qx * gx + qy * gy + qz * gz;
        float ad  = fminf(fabsf(dot), 1.0f - 1e-7f);
        rotv = acosf(ad) * cw;

        float p = fminf(fmaxf(class_pred[i], 1e-7f), 1.0f - 1e-7f);
        bcev = -(cw * logf(p) + (1.0f - cw) * logf(1.0f - p));
    }
    // full-wave sum via ds_swizzle xor exchanges
    transv = swz_add<SWZ_XOR(1)>(transv);  rotv = swz_add<SWZ_XOR(1)>(rotv);  bcev = swz_add<SWZ_XOR(1)>(bcev);
    transv = swz_add<SWZ_XOR(2)>(transv);  rotv = swz_add<SWZ_XOR(2)>(rotv);  bcev = swz_add<SWZ_XOR(2)>(bcev);
    transv = swz_add<SWZ_XOR(4)>(transv);  rotv = swz_add<SWZ_XOR(4)>(rotv);  bcev = swz_add<SWZ_XOR(4)>(bcev);
    transv = swz_add<SWZ_XOR(8)>(transv);  rotv = swz_add<SWZ_XOR(8)>(rotv);  bcev = swz_add<SWZ_XOR(8)>(bcev);
    transv = swz_add<SWZ_XOR(16)>(transv); rotv = swz_add<SWZ_XOR(16)>(rotv); bcev = swz_add<SWZ_XOR(16)>(bcev);
    if ((threadIdx.x & 31) == 0) {
        atomicAdd(&acc[6], transv);
        atomicAdd(&acc[7], rotv);
        atomicAdd(&acc[8], bcev);
    }
}

__global__ void finalize_kernel(const float* __restrict__ acc,
                                const float* __restrict__ temp,
                                float* __restrict__ out) {
    if (threadIdx.x == 0 && blockIdx.x == 0) {
        float meanSmp = acc[0] / (float)(BB * MM);
        const unsigned* mb = (const unsigned*)(acc + 1);
        float meanMax = 0.0f;
        #pragma unroll
        for (int b = 0; b < BB; ++b) meanMax += __uint_as_float(mb[b]);
        meanMax *= 0.25f;
        float meanOrg = acc[5] / (float)(BB * NN);
        float sample_loss = meanSmp + meanMax + GAMMA * meanOrg;
        float t = temp[0];
        float proj = t * t;
        float trans = acc[6] / (float)(BB * MM);
        float rot   = acc[7] / (float)(BB * MM);
        float reg = trans + ALPHA * rot;
        float cls = acc[8] / (float)(BB * MM);
        out[0] = THETA * sample_loss + proj + BETA * reg + cls;
    }
}

extern "C" void kernel_launch(void* const* d_in, const int* in_sizes, int n_in,
                              void* d_out, int out_size, void* d_ws, size_t ws_size,
                              hipStream_t stream) {
    const float* xyz        = (const float*)d_in[0];
    const float* sample_xyz = (const float*)d_in[1];
    const float* temp       = (const float*)d_in[2];
    const float* grasp_pred = (const float*)d_in[3];
    const float* grasp_gt   = (const float*)d_in[4];
    const float* class_pred = (const float*)d_in[5];
    const float* class_gt   = (const float*)d_in[6];
    float* acc = (float*)d_ws;
    float* out = (float*)d_out;

    init_acc_kernel<<<1, 32, 0, stream>>>(acc);

    {   // d_smp_org: sample_xyz vs xyz (needs per-batch max too)
        int tiles = BB * (MM >> 4);
        chamfer_min_kernel<<<(tiles + 7) / 8, 256, 0, stream>>>(
            sample_xyz, xyz, MM, NN, tiles, &acc[0], (unsigned*)&acc[1]);
    }
    {   // d_org_smp: xyz vs sample_xyz (sum only)
        int tiles = BB * (NN >> 4);
        chamfer_min_kernel<<<(tiles + 7) / 8, 256, 0, stream>>>(
            xyz, sample_xyz, NN, MM, tiles, &acc[5], nullptr);
    }
    grasp_cls_kernel<<<(BB * MM + 255) / 256, 256, 0, stream>>>(
        grasp_pred, grasp_gt, class_pred, class_gt, acc);
    finalize_kernel<<<1, 32, 0, stream>>>(acc, temp, out);
}